// KGAT_17265768530448
// MI455X (gfx1250) — compile-verified
//
#include <hip/hip_runtime.h>
#include <hip/hip_bf16.h>

typedef __attribute__((ext_vector_type(2))) float v2f;
typedef __attribute__((ext_vector_type(8))) float v8f;

#define NEG_SLOPE_F 0.01f
#define OUT_DIM 112

// ---------------------------------------------------------------------------
// Zero a float buffer (graph-capture-safe replacement for hipMemsetAsync).
__global__ void kgat_zero_kernel(float* __restrict__ p, long n) {
    long i = (long)blockIdx.x * blockDim.x + threadIdx.x;
    if (i < n) p[i] = 0.0f;
}

// ---------------------------------------------------------------------------
// Copy entity embeddings into columns [0,64) of the 112-wide output.
__global__ void kgat_copy_embed_kernel(const float* __restrict__ emb,
                                       float* __restrict__ outp, int n_nodes) {
    long i = (long)blockIdx.x * blockDim.x + threadIdx.x;
    if (i < (long)n_nodes * 64) {
        long n = i >> 6;            // /64
        int  j = (int)(i & 63);
        outp[n * OUT_DIM + j] = emb[i];
    }
}

// ---------------------------------------------------------------------------
// Precompute projected relations: relp[r][j] = sum_k rel[r][k]*pW[k][j] + pb[j]
// Only 40 x ID values; trivial.
__global__ void kgat_relproj_kernel(const float* __restrict__ rel,
                                    const float* __restrict__ pW,
                                    const float* __restrict__ pb,
                                    float* __restrict__ relp,
                                    int n_rel, int ID) {
    int idx = blockIdx.x * blockDim.x + threadIdx.x;
    if (idx >= n_rel * ID) return;
    int r = idx / ID, j = idx % ID;
    float s = pb[j];
    const float* rr = rel + (long)r * 64;
    for (int k = 0; k < 64; ++k) s = fmaf(rr[k], pW[k * ID + j], s);
    relp[idx] = s;
}

// ---------------------------------------------------------------------------
// Edge kernel: one wave32 per edge. Lane k handles feature dims k (and k+32
// when ID==64). Computes score = sum(h * tanh(h+r)) via wave reduction, then
// scatter-adds t*score onto neigh[head] with f32 global atomics.
template<int ID>
__global__ void kgat_edge_kernel(const float* __restrict__ cur, int cur_stride, int cur_off,
                                 const float* __restrict__ relp,
                                 const int* __restrict__ edge_index,
                                 const int* __restrict__ edge_type,
                                 float* __restrict__ neigh,
                                 int n_edges) {
    int lane = threadIdx.x & 31;
    int e = blockIdx.x * (blockDim.x >> 5) + (threadIdx.x >> 5);
    if (e >= n_edges) return;

    int head = edge_index[e];
    int tail = edge_index[n_edges + e];
    int rt   = edge_type[e];

    const float* hrow = cur + (long)head * cur_stride + cur_off;
    const float* trow = cur + (long)tail * cur_stride + cur_off;
    const float* rrow = relp + (long)rt * ID;

    float h0 = hrow[lane];
    float r0 = rrow[lane];
    float t0 = trow[lane];
    float s  = h0 * tanhf(h0 + r0);
    float t1 = 0.0f;
    if (ID == 64) {
        float h1 = hrow[lane + 32];
        float r1 = rrow[lane + 32];
        t1 = trow[lane + 32];
        s += h1 * tanhf(h1 + r1);
    }
    // wave32 all-reduce
    #pragma unroll
    for (int off = 16; off > 0; off >>= 1) s += __shfl_xor(s, off, 32);

    float* nrow = neigh + (long)head * ID;
    atomicAdd(&nrow[lane], t0 * s);
    if (ID == 64) atomicAdd(&nrow[lane + 32], t1 * s);
}

// ---------------------------------------------------------------------------
// Node kernel (WMMA): out = leaky_relu([cur+neigh | cur*neigh] @ [W1;W2] + b1+b2)
// then row L2-normalize. One wave per 16-node tile, 4 waves per block.
// Uses V_WMMA_F32_16X16X4_F32 accumulating over K = 2*ID in steps of 4.
// ISA 7.12.2 32-bit layouts:
//   A 16x4 : lanes 0-15 -> M=lane, {VGPR0,VGPR1} = K{0,1}; lanes 16-31 = K{2,3}
//   B 4x16 : lanes 0-15 -> N=lane, {VGPR0,VGPR1} = K{0,1}; lanes 16-31 = K{2,3}
//   D 16x16: VGPR v -> (M=v, N=lane) for lanes 0-15, (M=v+8, N=lane-16) else
template<int ID, int OD>
__global__ __launch_bounds__(128)
void kgat_node_kernel(const float* __restrict__ cur, int cur_stride, int cur_off,
                      const float* __restrict__ neigh,
                      const float* __restrict__ W1, const float* __restrict__ b1,
                      const float* __restrict__ W2, const float* __restrict__ b2,
                      float* __restrict__ outp, int out_off, int n_nodes) {
    constexpr int K = 2 * ID;
    constexpr int WAVES = 4;
    __shared__ float Atile[WAVES][16][K];
    __shared__ float Otile[WAVES][16][OD];
    __shared__ float Wlds[K][OD];
    __shared__ float blds[OD];

    const int tid  = threadIdx.x;
    const int lane = tid & 31;
    const int wv   = tid >> 5;

    // Stage fused weight matrix [W1;W2] and fused bias (whole block).
    for (int idx = tid; idx < ID * OD; idx += blockDim.x) {
        int r = idx / OD, c = idx % OD;
        Wlds[r][c]      = W1[idx];
        Wlds[r + ID][c] = W2[idx];
    }
    for (int idx = tid; idx < OD; idx += blockDim.x) blds[idx] = b1[idx] + b2[idx];

    const int nb = (blockIdx.x * WAVES + wv) * 16;  // base node of this wave's tile

    // Stage A = [cur+neigh | cur*neigh] for 16 nodes. OOB rows -> zeros.
    for (int idx = lane; idx < 16 * ID; idx += 32) {
        int row = idx / ID, col = idx % ID;
        int node = nb + row;
        float c = 0.0f, g = 0.0f;
        if (node < n_nodes) {
            c = cur[(long)node * cur_stride + cur_off + col];
            g = neigh[(long)node * ID + col];
        }
        Atile[wv][row][col]      = c + g;
        Atile[wv][row][col + ID] = c * g;
    }
    __syncthreads();

    const int row   = lane & 15;
    const int khalf = (lane >> 4) * 2;  // lanes 16-31 take K offsets +2,+3
    const int mbase = (lane >> 4) * 8;

    #pragma unroll
    for (int nt = 0; nt < OD / 16; ++nt) {
        v8f acc = {0.f, 0.f, 0.f, 0.f, 0.f, 0.f, 0.f, 0.f};
        #pragma unroll
        for (int kk = 0; kk < K / 4; ++kk) {
            const int kb = kk * 4 + khalf;
            v2f a, b;
            a.x = Atile[wv][row][kb];
            a.y = Atile[wv][row][kb + 1];
            b.x = Wlds[kb][nt * 16 + row];
            b.y = Wlds[kb + 1][nt * 16 + row];
            acc = __builtin_amdgcn_wmma_f32_16x16x4_f32(
                /*neg_a=*/false, a, /*neg_b=*/false, b,
                /*c_mod=*/(short)0, acc, /*reuse_a=*/false, /*reuse_b=*/false);
        }
        // Spill D tile to LDS with bias + leaky_relu.
        #pragma unroll
        for (int v = 0; v < 8; ++v) {
            int m   = mbase + v;
            int col = nt * 16 + row;
            float val = acc[v] + blds[col];
            val = val > 0.0f ? val : NEG_SLOPE_F * val;
            Otile[wv][m][col] = val;
        }
    }
    __syncthreads();

    // Row-wise L2 normalize and write into out[:, out_off : out_off+OD].
    if (lane < 16) {
        int node = nb + lane;
        if (node < n_nodes) {
            float ss = 0.0f;
            #pragma unroll
            for (int j = 0; j < OD; ++j) {
                float v = Otile[wv][lane][j];
                ss = fmaf(v, v, ss);
            }
            float scale = 1.0f / fmaxf(sqrtf(ss), 1e-12f);
            #pragma unroll
            for (int j = 0; j < OD; ++j)
                outp[(long)node * OUT_DIM + out_off + j] = Otile[wv][lane][j] * scale;
        }
    }
}

// ---------------------------------------------------------------------------
extern "C" void kernel_launch(void* const* d_in, const int* in_sizes, int n_in,
                              void* d_out, int out_size, void* d_ws, size_t ws_size,
                              hipStream_t stream) {
    const float* entity  = (const float*)d_in[0];   // [N,64]
    const float* relemb  = (const float*)d_in[1];   // [NREL,64]
    const float* projW0  = (const float*)d_in[2];   // [64,64]
    const float* projb0  = (const float*)d_in[3];
    const float* W1_0    = (const float*)d_in[4];   // [64,32]
    const float* b1_0    = (const float*)d_in[5];
    const float* W2_0    = (const float*)d_in[6];
    const float* b2_0    = (const float*)d_in[7];
    const float* projW1  = (const float*)d_in[8];   // [64,32]
    const float* projb1  = (const float*)d_in[9];
    const float* W1_1    = (const float*)d_in[10];  // [32,16]
    const float* b1_1    = (const float*)d_in[11];
    const float* W2_1    = (const float*)d_in[12];
    const float* b2_1    = (const float*)d_in[13];
    const int*   edge_ix = (const int*)d_in[14];    // [2,E]
    const int*   edge_ty = (const int*)d_in[15];    // [E]

    const int N    = in_sizes[0] / 64;
    const int NREL = in_sizes[1] / 64;
    const int E    = in_sizes[15];
    float* outF = (float*)d_out;

    // Workspace: neigh [N*64] | relp0 [NREL*64] | relp1 [NREL*32]
    float* neigh = (float*)d_ws;
    float* relp0 = neigh + (size_t)N * 64;
    float* relp1 = relp0 + (size_t)NREL * 64;

    const int T = 256;

    // Output columns [0,64): raw entity embeddings.
    {
        long tot = (long)N * 64;
        kgat_copy_embed_kernel<<<(unsigned)((tot + T - 1) / T), T, 0, stream>>>(entity, outF, N);
    }
    // Projected relations for both layers.
    kgat_relproj_kernel<<<(NREL * 64 + T - 1) / T, T, 0, stream>>>(relemb, projW0, projb0, relp0, NREL, 64);
    kgat_relproj_kernel<<<(NREL * 32 + T - 1) / T, T, 0, stream>>>(relemb, projW1, projb1, relp1, NREL, 32);

    const int edges_per_block = T / 32;
    const unsigned eblocks = (unsigned)((E + edges_per_block - 1) / edges_per_block);

    // ---- Layer 0: in=64 (entity, stride 64), out=32 -> out[:,64:96]
    {
        long tot = (long)N * 64;
        kgat_zero_kernel<<<(unsigned)((tot + T - 1) / T), T, 0, stream>>>(neigh, tot);
        kgat_edge_kernel<64><<<eblocks, T, 0, stream>>>(entity, 64, 0, relp0,
                                                        edge_ix, edge_ty, neigh, E);
        kgat_node_kernel<64, 32><<<(N + 63) / 64, 128, 0, stream>>>(
            entity, 64, 0, neigh, W1_0, b1_0, W2_0, b2_0, outF, /*out_off=*/64, N);
    }

    // ---- Layer 1: in=32 (read back from out[:,64:96], stride 112), out=16 -> out[:,96:112]
    {
        long tot = (long)N * 32;
        kgat_zero_kernel<<<(unsigned)((tot + T - 1) / T), T, 0, stream>>>(neigh, tot);
        kgat_edge_kernel<32><<<eblocks, T, 0, stream>>>(outF, OUT_DIM, 64, relp1,
                                                        edge_ix, edge_ty, neigh, E);
        kgat_node_kernel<32, 16><<<(N + 63) / 64, 128, 0, stream>>>(
            outF, OUT_DIM, 64, neigh, W1_1, b1_1, W2_1, b2_1, outF, /*out_off=*/96, N);
    }
}